// LogSlater2nd_43920335569135
// MI455X (gfx1250) — compile-verified
//
#include <hip/hip_runtime.h>

// ---------------------------------------------------------------------------
// Batched 32x32 log|det| (two sectors per sample) for MI455X / gfx1250.
//
// One wave32 per sample. Row-per-lane register LU with logical partial
// pivoting, blocked 16-wide: panel factor -> U12 triangular replay ->
// Schur complement A22' = A22 - L21*U12 via 4x v_wmma_f32_16x16x4_f32
// (chained over K) -> 16x16 trailing factorization.
//
// Output: complex64 per sample, flattened as float pairs (re, im).
//   re = sum log|u_kk|,  im = pi * ((#negative pivots + swap parity) & 1)
// ---------------------------------------------------------------------------

#define N_ORB 128
#define WAVES_PER_BLOCK 8
#define STR 20  // LDS tile row stride (floats): 80B rows -> 16B aligned, conflict-free

typedef float v2f __attribute__((ext_vector_type(2)));
typedef float v8f __attribute__((ext_vector_type(8)));

__device__ __forceinline__ float bcastf(float x, int srclane) {
  // wave-uniform broadcast via v_readlane -> SGPR (feeds VALU as scalar src)
  return __int_as_float(__builtin_amdgcn_readlane(__float_as_int(x), srclane));
}

__device__ __forceinline__ void wave_ds_fence() {
  // wave-local LDS store->load ordering (single wave owns its LDS region)
  __asm__ volatile("s_wait_dscnt 0x0" ::: "memory");
}

// One LU elimination step with logical pivoting, fully unrolled (K constant).
// Updates columns (K+1 .. JMAX-1). Stores the L multiplier into a[K] for
// lanes not yet chosen as pivots (needed by the U12 replay / WMMA Schur).
#define LUSTEP(K, JMAX)                                                        \
  do {                                                                         \
    unsigned key_ = (mystep == 64)                                             \
        ? ((__float_as_uint(a[(K)]) & 0x7FFFFFE0u) | (unsigned)lane)           \
        : 0u;                                                                  \
    _Pragma("unroll")                                                          \
    for (int off_ = 16; off_ > 0; off_ >>= 1) {                                \
      unsigned o_ = (unsigned)__shfl_xor((int)key_, off_, 32);                 \
      if (o_ > key_) key_ = o_;                                                \
    }                                                                          \
    int p_ = __builtin_amdgcn_readfirstlane((int)(key_ & 31u));                \
    float pval_ = bcastf(a[(K)], p_);                                          \
    sumlog += 0.69314718056f * __builtin_amdgcn_logf(__builtin_fabsf(pval_));  \
    neg += (int)(__float_as_uint(pval_) >> 31);                                \
    swaps += __popc(live & ((1u << p_) - 1u));                                 \
    live &= ~(1u << p_);                                                       \
    if (lane == p_) mystep = (K);                                              \
    float l_ = (mystep > (K)) ? a[(K)] * __builtin_amdgcn_rcpf(pval_) : 0.0f;  \
    float ln_ = -l_;                                                           \
    _Pragma("unroll")                                                          \
    for (int j_ = (K) + 1; j_ < (JMAX); ++j_) {                                \
      float u_ = bcastf(a[j_], p_);                                            \
      a[j_] = __builtin_fmaf(ln_, u_, a[j_]);                                  \
    }                                                                          \
    if (mystep > (K)) a[(K)] = l_;                                             \
    if ((K) < 16) pk[(K) & 15] = p_;                                           \
  } while (0)

__global__ __launch_bounds__(256, 1) void logslater_kernel(
    const int* __restrict__ n, const float* __restrict__ M0,
    const float* __restrict__ M1, float* __restrict__ out, int batch) {
  __shared__ int rowbuf[WAVES_PER_BLOCK * 32];
  __shared__ float bL[WAVES_PER_BLOCK * 16 * STR];  // -L21, reused for D
  __shared__ float bU[WAVES_PER_BLOCK * 16 * STR];  // U12
  __shared__ float bC[WAVES_PER_BLOCK * 16 * STR];  // A22 (C operand)

  const int lane = threadIdx.x & 31;
  const int wave = threadIdx.x >> 5;
  const int s = blockIdx.x * WAVES_PER_BLOCK + wave;
  if (s >= batch) return;

  int* rb = rowbuf + wave * 32;
  float* wL = bL + wave * 16 * STR;
  float* wU = bU + wave * 16 * STR;
  float* wC = bC + wave * 16 * STR;

  float re = 0.0f, im = 0.0f;

#pragma unroll 1
  for (int sec = 0; sec < 2; ++sec) {
    // ---- phase 0: occupied-orbital indices (rank + scatter via LDS) ----
    const int4 occ =
        ((const int4*)(n + (size_t)s * (2 * N_ORB) + sec * N_ORB))[lane];
    int cnt = occ.x + occ.y + occ.z + occ.w;
    int inc = cnt;
#pragma unroll
    for (int off = 1; off < 32; off <<= 1) {
      int t = __shfl_up(inc, off, 32);
      if (lane >= off) inc += t;
    }
    int pos = inc - cnt;  // exclusive prefix of set-bit counts
    if (occ.x) rb[pos++] = 4 * lane + 0;
    if (occ.y) rb[pos++] = 4 * lane + 1;
    if (occ.z) rb[pos++] = 4 * lane + 2;
    if (occ.w) rb[pos++] = 4 * lane + 3;
    wave_ds_fence();
    const int row = rb[lane];

    // ---- gather: lane r holds row r of A (32 fp32) ----
    const float* M = sec ? M1 : M0;
    float a[32];
    {
      const float4* mr = (const float4*)(M + (size_t)row * 32);
#pragma unroll
      for (int q = 0; q < 8; ++q) {
        float4 v = mr[q];
        a[4 * q + 0] = v.x;
        a[4 * q + 1] = v.y;
        a[4 * q + 2] = v.z;
        a[4 * q + 3] = v.w;
      }
    }

    int mystep = 64;  // step at which this lane was chosen as pivot (64 = not yet)
    int pk[16];       // pivot lane per panel step
    float sumlog = 0.0f;
    int neg = 0, swaps = 0;
    unsigned live = 0xFFFFFFFFu;

    // ---- phase 1: panel factorization, columns 0..15 only ----
#pragma unroll
    for (int k = 0; k < 16; ++k) {
      LUSTEP(k, 16);
    }

    // ---- phase 2a: U12 = L11^-1 * B1 (replay, pivot lanes only) ----
#pragma unroll
    for (int k = 0; k < 16; ++k) {
      float l2 = (mystep < 16 && mystep > k) ? a[k] : 0.0f;
      float ln2 = -l2;
#pragma unroll
      for (int j = 16; j < 32; ++j) {
        float u = bcastf(a[j], pk[k]);
        a[j] = __builtin_fmaf(ln2, u, a[j]);
      }
    }

    // ---- phase 2b: stage operands and Schur update via WMMA ----
    const bool isPiv = (mystep < 16);
    const unsigned nonpiv = ~((unsigned)__ballot(isPiv));
    if (isPiv) {  // U12 row (by elimination step)
      float* r = wU + mystep * STR;
#pragma unroll
      for (int q = 0; q < 4; ++q)
        *(float4*)(r + 4 * q) = make_float4(a[16 + 4 * q], a[17 + 4 * q],
                                            a[18 + 4 * q], a[19 + 4 * q]);
    } else {  // -L21 row and A22 row (by compact rank)
      int c = __popc(nonpiv & ((1u << lane) - 1u));
      float* rl = wL + c * STR;
      float* rc = wC + c * STR;
#pragma unroll
      for (int q = 0; q < 4; ++q) {
        *(float4*)(rl + 4 * q) = make_float4(-a[4 * q + 0], -a[4 * q + 1],
                                             -a[4 * q + 2], -a[4 * q + 3]);
        *(float4*)(rc + 4 * q) = make_float4(a[16 + 4 * q], a[17 + 4 * q],
                                             a[18 + 4 * q], a[19 + 4 * q]);
      }
    }
    wave_ds_fence();

    const int half = lane >> 4, li = lane & 15;
    // C/D layout: VGPR g, lane l -> row (g + 8*(l>=16)), col l%16
    v8f c8;
#pragma unroll
    for (int g = 0; g < 8; ++g) c8[g] = wC[(g + 8 * half) * STR + li];
    // A (16x4): lanes 0-15 M=lane {K,K+1}; lanes 16-31 M=lane-16 {K+2,K+3}
    // B (4x16): symmetric — VGPR0 rows {K, K+2}, VGPR1 rows {K+1, K+3}
    v2f Aop[4], Bop[4];
#pragma unroll
    for (int t = 0; t < 4; ++t) {
      Aop[t][0] = wL[li * STR + 4 * t + 2 * half + 0];
      Aop[t][1] = wL[li * STR + 4 * t + 2 * half + 1];
      Bop[t][0] = wU[(4 * t + 2 * half + 0) * STR + li];
      Bop[t][1] = wU[(4 * t + 2 * half + 1) * STR + li];
    }
#pragma unroll
    for (int t = 0; t < 4; ++t)
      c8 = __builtin_amdgcn_wmma_f32_16x16x4_f32(false, Aop[t], false, Bop[t],
                                                 (short)0, c8, false, false);
    // scatter D = A22 - L21*U12 back through LDS to the non-pivot lanes
#pragma unroll
    for (int g = 0; g < 8; ++g) wL[(g + 8 * half) * STR + li] = c8[g];
    wave_ds_fence();
    if (!isPiv) {
      int c = __popc(nonpiv & ((1u << lane) - 1u));
      const float* rd = wL + c * STR;
#pragma unroll
      for (int q = 0; q < 4; ++q) {
        float4 v = *(const float4*)(rd + 4 * q);
        a[16 + 4 * q] = v.x;
        a[17 + 4 * q] = v.y;
        a[18 + 4 * q] = v.z;
        a[19 + 4 * q] = v.w;
      }
    }

    // ---- phase 3: factor the 16x16 Schur block (columns 16..31) ----
#pragma unroll
    for (int k = 16; k < 32; ++k) {
      LUSTEP(k, 32);
    }

    re += sumlog;
    im += 3.14159265358979f * (float)((neg + swaps) & 1);
  }

  if (lane == 0) {
    float2 o;
    o.x = re;
    o.y = im;
    reinterpret_cast<float2*>(out)[s] = o;
  }
}

extern "C" void kernel_launch(void* const* d_in, const int* in_sizes, int n_in,
                              void* d_out, int out_size, void* d_ws,
                              size_t ws_size, hipStream_t stream) {
  (void)n_in;
  (void)out_size;
  (void)d_ws;
  (void)ws_size;
  const int* n = (const int*)d_in[0];
  const float* M0 = (const float*)d_in[1];
  const float* M1 = (const float*)d_in[2];
  float* out = (float*)d_out;
  const int batch = in_sizes[0] / (2 * N_ORB);  // 65536
  const int blocks = (batch + WAVES_PER_BLOCK - 1) / WAVES_PER_BLOCK;
  logslater_kernel<<<blocks, 32 * WAVES_PER_BLOCK, 0, stream>>>(n, M0, M1, out,
                                                                batch);
}